// LSInput_79001628443218
// MI455X (gfx1250) — compile-verified
//
#include <hip/hip_runtime.h>
#include <hip/hip_bf16.h>

// ---------------------------------------------------------------------------
// MI455X / gfx1250 implementation.
//  - All matrix math uses V_WMMA_F32_16X16X4_F32 (fp32 matrix pipe, wave32).
//  - Weights (64KB) + gathered activation rows staged in LDS (320KB/WGP).
//  - Edge kernel keeps B-fragments register-resident: 256B/wave LDS traffic
//    per WMMA (the floor), matching LDS peak for 4 SIMDs/WGP.
//  - Edge scatter via global_atomic_add_f32 (order-independent +=).
// ---------------------------------------------------------------------------

typedef float v2f __attribute__((ext_vector_type(2)));
typedef float v8f __attribute__((ext_vector_type(8)));

#define CC   128          // channels
#define APAD 132          // LDS stride for activation tiles (conflict-free A reads)
#define WPAD 136          // LDS stride for weight tiles     (conflict-free B reads)
#define GN_EPS 1e-5f

__device__ __forceinline__ v8f wmma_f32(v2f a, v2f b, v8f c) {
  // 8 args: (neg_a, A, neg_b, B, c_mod, C, reuse_a, reuse_b)
  return __builtin_amdgcn_wmma_f32_16x16x4_f32(false, a, false, b, (short)0, c,
                                               false, false);
}

// ---------------------------------------------------------------------------
// Stem: h1 = relu(ctrs @ W_in1 + b_in1), h2 = relu(feats @ W_seg1 + b_seg1)
// ---------------------------------------------------------------------------
__global__ __launch_bounds__(256)
void stem_kernel(const float* __restrict__ ctrs, const float* __restrict__ feats,
                 const float* __restrict__ W1a, const float* __restrict__ b1a,
                 const float* __restrict__ W1b, const float* __restrict__ b1b,
                 float* __restrict__ h1, float* __restrict__ h2, int total) {
  int idx = blockIdx.x * 256 + threadIdx.x;
  if (idx >= total) return;
  int n = idx >> 7;          // / CC
  int c = idx & (CC - 1);
  float c0 = ctrs[2 * n], c1 = ctrs[2 * n + 1];
  float f0 = feats[2 * n], f1 = feats[2 * n + 1];
  h1[idx] = fmaxf(c0 * W1a[c] + c1 * W1a[CC + c] + b1a[c], 0.0f);
  h2[idx] = fmaxf(f0 * W1b[c] + f1 * W1b[CC + c] + b1b[c], 0.0f);
}

// ---------------------------------------------------------------------------
// Fused GEMM (out = A @ W) + GroupNorm(1 group over CC) [+ addend + relu].
// One workgroup = 16 rows x 128 cols; 8 waves, one 16x16 col-tile each.
// mode 0: out = gn(A@W)
// mode 1: out = relu(gn(A@W) + addend); if out2, also write out2 = out
// ---------------------------------------------------------------------------
__global__ __launch_bounds__(256)
void gemm_gn_kernel(const float* __restrict__ A, const float* __restrict__ W,
                    const float* __restrict__ g, const float* __restrict__ bb,
                    const float* __restrict__ addend, float* __restrict__ out,
                    float* __restrict__ out2, int N, int mode) {
  __shared__ float Wl[CC * WPAD];
  __shared__ float Al[16 * APAD];
  __shared__ float Ol[16 * APAD];
  __shared__ float stats[32];

  const int tid  = threadIdx.x;
  const int wave = tid >> 5;
  const int lane = tid & 31;
  const int half = lane >> 4;
  const int l    = lane & 15;
  const long rowBase = (long)blockIdx.x * 16;

  // Stage weight matrix (64KB) in LDS.
  for (int idx = tid; idx < CC * CC; idx += 256) {
    int k = idx >> 7, c = idx & (CC - 1);
    Wl[k * WPAD + c] = W[idx];
  }
  // Stage 16 activation rows (8KB) in LDS (float4, coalesced).
  for (int q = 0; q < 2; ++q) {
    int vi = tid + q * 256;              // 512 float4 total
    int r  = vi >> 5;                    // (vi*4)/128
    int c4 = (vi & 31) * 4;
    long row = rowBase + r;
    if (row > (long)N - 1) row = (long)N - 1;
    float4 d = *(const float4*)(A + row * CC + c4);
    *(float4*)&Al[r * APAD + c4] = d;
  }
  __syncthreads();

  // WMMA k-loop: K=128 in steps of 4.
  const int cb = wave * 16;
  v8f acc = {};
#pragma unroll
  for (int k = 0; k < CC; k += 4) {
    v2f a, b;
    a.x = Al[l * APAD + k + 2 * half];
    a.y = Al[l * APAD + k + 2 * half + 1];
    b.x = Wl[(k + 2 * half) * WPAD + cb + l];
    b.y = Wl[(k + 2 * half + 1) * WPAD + cb + l];
    acc = wmma_f32(a, b, acc);
  }
  // D tile -> LDS (row = j + 8*half, col = cb + l).
#pragma unroll
  for (int j = 0; j < 8; ++j)
    Ol[(j + 8 * half) * APAD + cb + l] = acc[j];
  __syncthreads();

  // GroupNorm stats: 16 threads per row, shfl-reduce within width-16 groups.
  const int r  = tid >> 4;
  const int jj = tid & 15;
  float s = 0.f, s2 = 0.f;
#pragma unroll
  for (int q = 0; q < 8; ++q) {
    float v = Ol[r * APAD + jj * 8 + q];
    s += v; s2 += v * v;
  }
  for (int off = 8; off >= 1; off >>= 1) {
    s  += __shfl_xor(s, off, 16);
    s2 += __shfl_xor(s2, off, 16);
  }
  if (jj == 0) { stats[r * 2] = s * (1.0f / CC); stats[r * 2 + 1] = s2 * (1.0f / CC); }
  __syncthreads();

  const float m   = stats[r * 2];
  const float var = stats[r * 2 + 1] - m * m;
  const float inv = rsqrtf(var + GN_EPS);
  const long grow = rowBase + r;
  if (grow < N) {
#pragma unroll
    for (int q = 0; q < 8; ++q) {
      int c = jj * 8 + q;
      float v = (Ol[r * APAD + c] - m) * inv * g[c] + bb[c];
      if (mode == 1) {
        v += addend[grow * CC + c];
        v = fmaxf(v, 0.0f);
      }
      out[grow * CC + c] = v;
      if (out2) out2[grow * CC + c] = v;
    }
  }
}

// ---------------------------------------------------------------------------
// Plain GEMM: out = A @ W  (temp = feat @ W_ctr[i])
// ---------------------------------------------------------------------------
__global__ __launch_bounds__(256)
void gemm_raw_kernel(const float* __restrict__ A, const float* __restrict__ W,
                     float* __restrict__ out, int N) {
  __shared__ float Wl[CC * WPAD];
  __shared__ float Al[16 * APAD];

  const int tid  = threadIdx.x;
  const int wave = tid >> 5;
  const int lane = tid & 31;
  const int half = lane >> 4;
  const int l    = lane & 15;
  const long rowBase = (long)blockIdx.x * 16;

  for (int idx = tid; idx < CC * CC; idx += 256) {
    int k = idx >> 7, c = idx & (CC - 1);
    Wl[k * WPAD + c] = W[idx];
  }
  for (int q = 0; q < 2; ++q) {
    int vi = tid + q * 256;
    int r  = vi >> 5;
    int c4 = (vi & 31) * 4;
    long row = rowBase + r;
    if (row > (long)N - 1) row = (long)N - 1;
    float4 d = *(const float4*)(A + row * CC + c4);
    *(float4*)&Al[r * APAD + c4] = d;
  }
  __syncthreads();

  const int cb = wave * 16;
  v8f acc = {};
#pragma unroll
  for (int k = 0; k < CC; k += 4) {
    v2f a, b;
    a.x = Al[l * APAD + k + 2 * half];
    a.y = Al[l * APAD + k + 2 * half + 1];
    b.x = Wl[(k + 2 * half) * WPAD + cb + l];
    b.y = Wl[(k + 2 * half + 1) * WPAD + cb + l];
    acc = wmma_f32(a, b, acc);
  }
#pragma unroll
  for (int j = 0; j < 8; ++j) {
    long grow = rowBase + j + 8 * half;
    if (grow < N) out[grow * CC + cb + l] = acc[j];
  }
}

// ---------------------------------------------------------------------------
// Edge scatter: temp[u[e]] += feat[v[e]] @ W, for all 14 edge sets (grid.y).
// WG = 128 edges. Wave w owns column tile w for ALL 128 edges: its 64 weight
// fragment floats are loaded from LDS once into registers and reused across
// 8 edge groups -> only the A pair (8B/lane) hits LDS per WMMA.
// ---------------------------------------------------------------------------
__global__ __launch_bounds__(256)
void edge_kernel(const float* __restrict__ feat, float* __restrict__ temp,
                 const int* __restrict__ pre_u, const int* __restrict__ pre_v,
                 const int* __restrict__ suc_u, const int* __restrict__ suc_v,
                 const int* __restrict__ left_u, const int* __restrict__ left_v,
                 const int* __restrict__ right_u, const int* __restrict__ right_v,
                 const float* __restrict__ W_pre, const float* __restrict__ W_suc,
                 const float* __restrict__ W_left, const float* __restrict__ W_right,
                 int blk, int S, int E, int E2) {
  const int set = blockIdx.y;
  const int* u; const int* v; const float* W; long n;
  if (set < S) {
    u = pre_u + (size_t)set * E;  v = pre_v + (size_t)set * E;
    W = W_pre + ((size_t)blk * S + set) * CC * CC;  n = E;
  } else if (set < 2 * S) {
    int s = set - S;
    u = suc_u + (size_t)s * E;    v = suc_v + (size_t)s * E;
    W = W_suc + ((size_t)blk * S + s) * CC * CC;    n = E;
  } else if (set == 2 * S) {
    u = left_u;  v = left_v;  W = W_left  + (size_t)blk * CC * CC;  n = E2;
  } else {
    u = right_u; v = right_v; W = W_right + (size_t)blk * CC * CC;  n = E2;
  }
  const long base = (long)blockIdx.x * 128;
  if (base >= n) return;   // uniform early exit (whole workgroup)

  __shared__ float Wl[CC * WPAD];   // ~69.6 KB
  __shared__ float Al[128 * APAD];  // ~67.6 KB  (2 WGs / 320KB WGP)
  __shared__ int   Ul[128];         // scatter row indices (broadcast reads)

  const int tid = threadIdx.x;
  for (int idx = tid; idx < CC * CC; idx += 256) {
    int k = idx >> 7, c = idx & (CC - 1);
    Wl[k * WPAD + c] = W[idx];
  }
  if (tid < 128) {
    long e = base + tid;
    Ul[tid] = (e < n) ? u[e] : 0;
  }
  // Gather 128 rows of feat (random rows; feat ~fits MI455X 192MB L2).
  {
    const int el    = tid >> 1;      // edge-local row 0..127
    const int hcol  = (tid & 1) * 64;
    long e = base + el;
    int row = (e < n) ? v[e] : 0;    // clamp: results discarded at scatter
    const float4* src = (const float4*)(feat + (size_t)row * CC + hcol);
#pragma unroll
    for (int q = 0; q < 16; ++q) {
      float4 d = src[q];
      *(float4*)&Al[el * APAD + hcol + q * 4] = d;
    }
  }
  __syncthreads();

  const int wave = tid >> 5;
  const int lane = tid & 31;
  const int half = lane >> 4;
  const int l    = lane & 15;
  const int cb   = wave * 16;       // this wave's column tile (all 128 edges)

  // Preload this wave's 32 B-fragments (64 floats/lane) into registers.
  v2f breg[32];
#pragma unroll
  for (int kk = 0; kk < 32; ++kk) {
    breg[kk].x = Wl[(4 * kk + 2 * half) * WPAD + cb + l];
    breg[kk].y = Wl[(4 * kk + 2 * half + 1) * WPAD + cb + l];
  }

  for (int gr = 0; gr < 8; ++gr) {  // 8 groups of 16 edges
    const int erow = gr * 16 + l;
    v8f acc = {};
#pragma unroll
    for (int kk = 0; kk < 32; ++kk) {
      v2f a;
      a.x = Al[erow * APAD + 4 * kk + 2 * half];
      a.y = Al[erow * APAD + 4 * kk + 2 * half + 1];
      acc = wmma_f32(a, breg[kk], acc);
    }
#pragma unroll
    for (int j = 0; j < 8; ++j) {
      int elocal = gr * 16 + j + 8 * half;
      long e = base + elocal;
      if (e < n) {
        int ur = Ul[elocal];        // LDS broadcast (same addr, lanes 0-15)
        atomicAdd(&temp[(size_t)ur * CC + cb + l], acc[j]);
      }
    }
  }
}

// ---------------------------------------------------------------------------
// Standalone row GroupNorm + ReLU, in place: x = relu(gn(x, g, b))
// ---------------------------------------------------------------------------
__global__ __launch_bounds__(256)
void gn_relu_kernel(float* __restrict__ x, const float* __restrict__ g,
                    const float* __restrict__ bb, int N) {
  __shared__ float stats[32];
  const int tid = threadIdx.x;
  const int r   = tid >> 4;
  const int jj  = tid & 15;
  const long grow = (long)blockIdx.x * 16 + r;
  float vals[8];
  float s = 0.f, s2 = 0.f;
  if (grow < N) {
    float4 a = *(const float4*)(x + grow * CC + jj * 8);
    float4 b = *(const float4*)(x + grow * CC + jj * 8 + 4);
    vals[0] = a.x; vals[1] = a.y; vals[2] = a.z; vals[3] = a.w;
    vals[4] = b.x; vals[5] = b.y; vals[6] = b.z; vals[7] = b.w;
#pragma unroll
    for (int q = 0; q < 8; ++q) { s += vals[q]; s2 += vals[q] * vals[q]; }
  } else {
#pragma unroll
    for (int q = 0; q < 8; ++q) vals[q] = 0.f;
  }
  for (int off = 8; off >= 1; off >>= 1) {
    s  += __shfl_xor(s, off, 16);
    s2 += __shfl_xor(s2, off, 16);
  }
  if (jj == 0) { stats[r * 2] = s * (1.0f / CC); stats[r * 2 + 1] = s2 * (1.0f / CC); }
  __syncthreads();
  if (grow < N) {
    const float m   = stats[r * 2];
    const float inv = rsqrtf(stats[r * 2 + 1] - m * m + GN_EPS);
#pragma unroll
    for (int q = 0; q < 8; ++q) {
      int c = jj * 8 + q;
      float v = (vals[q] - m) * inv * g[c] + bb[c];
      x[grow * CC + c] = fmaxf(v, 0.0f);
    }
  }
}

// ---------------------------------------------------------------------------
extern "C" void kernel_launch(void* const* d_in, const int* in_sizes, int n_in,
                              void* d_out, int out_size, void* d_ws, size_t ws_size,
                              hipStream_t stream) {
  const float* ctrs   = (const float*)d_in[0];
  const float* feats  = (const float*)d_in[1];
  const int* pre_u    = (const int*)d_in[2];
  const int* pre_v    = (const int*)d_in[3];
  const int* suc_u    = (const int*)d_in[4];
  const int* suc_v    = (const int*)d_in[5];
  const int* left_u   = (const int*)d_in[6];
  const int* left_v   = (const int*)d_in[7];
  const int* right_u  = (const int*)d_in[8];
  const int* right_v  = (const int*)d_in[9];
  const float* W_in1  = (const float*)d_in[10];
  const float* b_in1  = (const float*)d_in[11];
  const float* W_in2  = (const float*)d_in[12];
  const float* g_in   = (const float*)d_in[13];
  const float* bb_in  = (const float*)d_in[14];
  const float* W_seg1 = (const float*)d_in[15];
  const float* b_seg1 = (const float*)d_in[16];
  const float* W_seg2 = (const float*)d_in[17];
  const float* g_seg  = (const float*)d_in[18];
  const float* bb_seg = (const float*)d_in[19];
  const float* W_ctr  = (const float*)d_in[20];
  const float* W_pre  = (const float*)d_in[21];
  const float* W_suc  = (const float*)d_in[22];
  const float* W_left = (const float*)d_in[23];
  const float* W_right= (const float*)d_in[24];
  const float* g_norm = (const float*)d_in[25];
  const float* b_norm = (const float*)d_in[26];
  const float* W_ctr2 = (const float*)d_in[27];
  const float* g_ctr2 = (const float*)d_in[28];
  const float* b_ctr2 = (const float*)d_in[29];

  const int N  = in_sizes[0] / 2;
  const int E2 = in_sizes[6];
  const int B  = in_sizes[20] / (CC * CC);
  const int S  = in_sizes[21] / (B * CC * CC);
  const int E  = in_sizes[2] / S;
  const size_t NC = (size_t)N * CC;

  float* B0 = (float*)d_ws;          // res / stem-in scratch
  float* B1 = B0 + NC;               // feat ping
  float* B2 = B1 + NC;               // temp
  float* OUT = (float*)d_out;        // feat pong + final output

  const int rowBlocks = (N + 15) / 16;
  const int total = N * CC;

  // Stems: B0 = relu(ctrs@W_in1+b), B1 = relu(feats@W_seg1+b)
  stem_kernel<<<(total + 255) / 256, 256, 0, stream>>>(
      ctrs, feats, W_in1, b_in1, W_seg1, b_seg1, B0, B1, total);
  // B0 = gn(B0 @ W_in2)            (in place: rows staged to LDS first)
  gemm_gn_kernel<<<rowBlocks, 256, 0, stream>>>(
      B0, W_in2, g_in, bb_in, nullptr, B0, nullptr, N, 0);
  // B1 = relu(gn(B1 @ W_seg2) + B0); B0 = B1 (res)
  gemm_gn_kernel<<<rowBlocks, 256, 0, stream>>>(
      B1, W_seg2, g_seg, bb_seg, B0, B1, B0, N, 1);

  float* feat = B1;
  const int Emax = (E > E2) ? E : E2;
  dim3 egrid((Emax + 127) / 128, 2 * S + 2);

  for (int i = 0; i < B; ++i) {
    // temp = feat @ W_ctr[i]
    gemm_raw_kernel<<<rowBlocks, 256, 0, stream>>>(
        feat, W_ctr + (size_t)i * CC * CC, B2, N);
    // temp[u] += feat[v] @ W  for all 14 edge sets
    edge_kernel<<<egrid, 256, 0, stream>>>(
        feat, B2, pre_u, pre_v, suc_u, suc_v, left_u, left_v, right_u, right_v,
        W_pre, W_suc, W_left, W_right, i, S, E, E2);
    // temp = relu(gn(temp))
    gn_relu_kernel<<<rowBlocks, 256, 0, stream>>>(
        B2, g_norm + (size_t)i * CC, b_norm + (size_t)i * CC, N);
    // feat' = relu(gn(temp @ W_ctr2[i]) + res); res = feat'
    float* nxt = (i == B - 1) ? OUT : ((feat == B1) ? OUT : B1);
    gemm_gn_kernel<<<rowBlocks, 256, 0, stream>>>(
        B2, W_ctr2 + (size_t)i * CC * CC, g_ctr2 + (size_t)i * CC,
        b_ctr2 + (size_t)i * CC, B0, nxt, B0, N, 1);
    feat = nxt;
  }
}